// Model2d3d_16982300688817
// MI455X (gfx1250) — compile-verified
//
#include <hip/hip_runtime.h>

// ---------------------------------------------------------------------------
// Model2d3d forward for MI455X (gfx1250, wave32, WMMA).
// All GEMMs use V_WMMA_F32_16X16X4_F32 (fp32 in / fp32 acc, reference-exact).
// One wave per 16(M)x64(N) strip -> 4 independent WMMA accumulator chains.
// Inner K-loops are guard-free (main loop over K&~3 with pointer increments);
// K tails use clamped addresses + value selects; the fused gather layer's
// xyz/feature split is handled by ADDRESS selects (v_cndmask), so the whole
// kernel is free of EXEC divergence.
// ---------------------------------------------------------------------------

#define B_    8
#define V_    3
#define CIMG  128
#define P_    4096
#define HW_   1312            // 32*41
#define NCLS  21
#define RSQ   0.9999950000374997f   // 1/sqrt(1+1e-5)

typedef __attribute__((ext_vector_type(2))) float v2f;
typedef __attribute__((ext_vector_type(8))) float v8f;

__device__ __forceinline__ v8f wmma_f32_k4(v2f a, v2f b, v8f c) {
  // D(16x16,f32) = A(16x4,f32) x B(4x16,f32) + C
  return __builtin_amdgcn_wmma_f32_16x16x4_f32(
      /*neg_a=*/false, a, /*neg_b=*/false, b,
      /*c_mod=*/(short)0, c, /*reuse_a=*/false, /*reuse_b=*/false);
}

// ---------------------------------------------------------------------------
// Step 0: scatter projection indices: src_of[bv][pos=i3[k]] = i2[k], k=1..P
// ---------------------------------------------------------------------------
__global__ void k_scatter(const int* __restrict__ i3, const int* __restrict__ i2,
                          int* __restrict__ src_of) {
  int t = blockIdx.x * blockDim.x + threadIdx.x;
  if (t >= B_ * V_ * P_) return;
  int bv = t / P_, k = t % P_;
  int pos = i3[bv * (P_ + 1) + k + 1];
  int src = i2[bv * (P_ + 1) + k + 1];
  src_of[bv * P_ + pos] = src;
}

// ---------------------------------------------------------------------------
// Step 1: imf[b][c][p] = max over views v of img[(b*V+v)][c][src_of[bv][p]]
// ---------------------------------------------------------------------------
__global__ void k_imf(const float* __restrict__ img, const int* __restrict__ src_of,
                      float* __restrict__ feat0) {
  int t = blockIdx.x * blockDim.x + threadIdx.x;
  if (t >= B_ * CIMG * P_) return;
  int p = t % P_;
  int c = (t / P_) % CIMG;
  int b = t / (P_ * CIMG);
  float m = -3.4e38f;
  for (int v = 0; v < V_; ++v) {
    int bv = b * V_ + v;
    int s  = src_of[(size_t)bv * P_ + p];
    float val = img[((size_t)bv * CIMG + c) * HW_ + s];
    m = fmaxf(m, val);
  }
  feat0[t] = m;
}

// ---------------------------------------------------------------------------
// Farthest point sampling: one block (256 threads) per batch.
// ---------------------------------------------------------------------------
__global__ void k_fps(const float* __restrict__ xyz, int N, int npoint,
                      int* __restrict__ outidx) {
  __shared__ float dists[4096];
  __shared__ float rv[256];
  __shared__ int   ri[256];
  int b = blockIdx.x;
  const float* x = xyz + (size_t)b * N * 3;
  for (int i = threadIdx.x; i < N; i += 256) dists[i] = 1e10f;
  __syncthreads();
  int last = 0;
  for (int it = 0; it < npoint; ++it) {
    if (threadIdx.x == 0) outidx[b * npoint + it] = last;
    float lx = x[last * 3 + 0], ly = x[last * 3 + 1], lz = x[last * 3 + 2];
    float best = -1.0f; int bi = 0;
    for (int i = threadIdx.x; i < N; i += 256) {
      float dx = x[i * 3 + 0] - lx, dy = x[i * 3 + 1] - ly, dz = x[i * 3 + 2] - lz;
      float d  = dx * dx + dy * dy + dz * dz;
      float nd = fminf(dists[i], d);
      dists[i] = nd;
      if (nd > best) { best = nd; bi = i; }   // strict > keeps first index
    }
    rv[threadIdx.x] = best; ri[threadIdx.x] = bi;
    __syncthreads();
    for (int s = 128; s > 0; s >>= 1) {
      if (threadIdx.x < s) {
        float ov = rv[threadIdx.x + s]; int oi = ri[threadIdx.x + s];
        float cv = rv[threadIdx.x];     int ci = ri[threadIdx.x];
        if (ov > cv || (ov == cv && oi < ci)) { rv[threadIdx.x] = ov; ri[threadIdx.x] = oi; }
      }
      __syncthreads();
    }
    last = ri[0];
    __syncthreads();
  }
}

__global__ void k_gather_xyz(const float* __restrict__ xyz, const int* __restrict__ idx,
                             int N, int np, float* __restrict__ nxyz) {
  int t = blockIdx.x * blockDim.x + threadIdx.x;
  if (t >= B_ * np) return;
  int b = t / np;
  int s = idx[t];
  const float* src = xyz + ((size_t)b * N + s) * 3;
  float* dst = nxyz + (size_t)t * 3;
  dst[0] = src[0]; dst[1] = src[1]; dst[2] = src[2];
}

// ---------------------------------------------------------------------------
// Ball query: first 32 in-radius indices ascending, padded with first hit.
// ---------------------------------------------------------------------------
__global__ void k_ball(const float* __restrict__ xyz, const float* __restrict__ nxyz,
                       int N, int np, float r2, int* __restrict__ bidx) {
  int t = blockIdx.x * blockDim.x + threadIdx.x;
  if (t >= B_ * np) return;
  int b = t / np;
  const float* q = nxyz + (size_t)t * 3;
  float cx = q[0], cy = q[1], cz = q[2];
  const float* x = xyz + (size_t)b * N * 3;
  int* out = bidx + (size_t)t * 32;
  int cnt = 0, first = 0;
  for (int i = 0; i < N && cnt < 32; ++i) {
    float dx = x[i * 3 + 0] - cx, dy = x[i * 3 + 1] - cy, dz = x[i * 3 + 2] - cz;
    float d2 = dx * dx + dy * dy + dz * dz;
    if (d2 < r2) { if (cnt == 0) first = i; out[cnt++] = i; }
  }
  int pad = (cnt > 0) ? first : 0;
  for (; cnt < 32; ++cnt) out[cnt] = pad;
}

// ---------------------------------------------------------------------------
// SA layer-1 GEMM fused with ball-query gather.
// k0=0 peeled: address-selected gathers (branchless); main loop = pure
// feature-row gathers, guard-free; clamp+select K tail.
// grid = (np*32/64, M/16, B), block = 32 (one wave).
// ---------------------------------------------------------------------------
__global__ void k_sa_gemm1(const float* __restrict__ xyzp, const float* __restrict__ nxyz,
                           const float* __restrict__ featp, const int* __restrict__ bidx,
                           const float* __restrict__ W, const float* __restrict__ gam,
                           const float* __restrict__ bet, float* __restrict__ out,
                           int N, int np, int Cf, int M) {
  const int K    = 3 + Cf;
  const int Ncol = np * 32;
  int bz   = blockIdx.z;
  int m0   = blockIdx.y * 16;
  int n0   = blockIdx.x * 64;
  int lane = threadIdx.x;
  bool hi  = (lane >> 4) != 0;        // lane half
  int half = hi ? 1 : 0;
  int li   = lane & 15;

  int   pidx[4];
  float nx0[4], nx1[4], nx2[4];
#pragma unroll
  for (int t = 0; t < 4; ++t) {
    int col = n0 + t * 16 + li;
    int g = col >> 5, s = col & 31;
    pidx[t] = bidx[((size_t)bz * np + g) * 32 + s];
    const float* q = nxyz + ((size_t)bz * np + g) * 3;
    nx0[t] = q[0]; nx1[t] = q[1]; nx2[t] = q[2];
  }
  const float* Wrow = W + (size_t)(m0 + li) * K;   // M always multiple of 16 here
  const float* fb   = featp + (size_t)bz * Cf * N;
  const float* xb   = xyzp + (size_t)bz * N * 3;
  const float* wa   = Wrow + 2 * half;

  v8f acc[4] = {};

  // ---- peeled k0 = 0: rows {0,1,2}=centered xyz, row 3 = feature row 0.
  // Select ADDRESSES per lane-half (v_cndmask), loads stay unconditional.
  {
    v2f a; a[0] = wa[0]; a[1] = wa[1];
#pragma unroll
    for (int t = 0; t < 4; ++t) {
      const float* addr0 = xb + (size_t)pidx[t] * 3 + 2 * half;            // x | z
      const float* addr1 = hi ? (fb + pidx[t])
                              : (xb + (size_t)pidx[t] * 3 + 1);            // f0 | y
      float sub0 = hi ? nx2[t] : nx0[t];
      float sub1 = hi ? 0.0f   : nx1[t];
      v2f bf;
      bf[0] = addr0[0] - sub0;
      bf[1] = addr1[0] - sub1;
      acc[t] = wmma_f32_k4(a, bf, acc[t]);
    }
    wa += 4;
  }

  // ---- main loop: kk in [4, k0end), feature rows fr = kk-3 >= 1 ----
  const float* fq0 = fb + (size_t)(1 + 2 * half) * N + pidx[0];
  const float* fq1 = fb + (size_t)(1 + 2 * half) * N + pidx[1];
  const float* fq2 = fb + (size_t)(1 + 2 * half) * N + pidx[2];
  const float* fq3 = fb + (size_t)(1 + 2 * half) * N + pidx[3];
  int k0 = 4;
  for (; k0 + 4 <= K; k0 += 4) {
    v2f a; a[0] = wa[0]; a[1] = wa[1];
    v2f q0; q0[0] = fq0[0]; q0[1] = fq0[N];
    v2f q1; q1[0] = fq1[0]; q1[1] = fq1[N];
    v2f q2; q2[0] = fq2[0]; q2[1] = fq2[N];
    v2f q3; q3[0] = fq3[0]; q3[1] = fq3[N];
    acc[0] = wmma_f32_k4(a, q0, acc[0]);
    acc[1] = wmma_f32_k4(a, q1, acc[1]);
    acc[2] = wmma_f32_k4(a, q2, acc[2]);
    acc[3] = wmma_f32_k4(a, q3, acc[3]);
    wa += 4;
    fq0 += (size_t)4 * N; fq1 += (size_t)4 * N;
    fq2 += (size_t)4 * N; fq3 += (size_t)4 * N;
  }

  // ---- tail: 1-3 leftover rows, clamped addresses + zero-selected B ----
  if (k0 < K) {
    int ka  = k0 + 2 * half;
    int c0  = (ka     < K) ? ka     : K - 1;
    int c1  = (ka + 1 < K) ? ka + 1 : K - 1;
    bool ok0 = ka < K, ok1 = (ka + 1) < K;
    v2f a; a[0] = Wrow[c0]; a[1] = Wrow[c1];
    size_t r0 = (size_t)(c0 - 3) * N, r1 = (size_t)(c1 - 3) * N;
#pragma unroll
    for (int t = 0; t < 4; ++t) {
      float v0 = fb[r0 + pidx[t]];
      float v1 = fb[r1 + pidx[t]];
      v2f bf;
      bf[0] = ok0 ? v0 : 0.0f;
      bf[1] = ok1 ? v1 : 0.0f;
      acc[t] = wmma_f32_k4(a, bf, acc[t]);
    }
  }

#pragma unroll
  for (int t = 0; t < 4; ++t) {
    int col = n0 + t * 16 + li;
#pragma unroll
    for (int j = 0; j < 8; ++j) {
      int m = m0 + j + 8 * half;
      float v = acc[t][j] * RSQ * gam[m] + bet[m];
      v = fmaxf(v, 0.0f);
      out[((size_t)bz * M + m) * Ncol + col] = v;
    }
  }
}

// ---------------------------------------------------------------------------
// Generic batched GEMM: Y[b][M][N] = op(W[M][K] x X[b][K][N]).
// Guard-free main loop over K&~3; clamp+select tail; M tail clamped at the
// A address (never stored) and guarded only at the store.
// ---------------------------------------------------------------------------
__global__ void k_gemm(const float* __restrict__ X, const float* __restrict__ W,
                       const float* __restrict__ gam, const float* __restrict__ bet,
                       float* __restrict__ Y, int M, int N, int K, int relu) {
  int bz = blockIdx.z;
  const float* Xb = X + (size_t)bz * K * N;
  float*       Yb = Y + (size_t)bz * M * N;
  int m0 = blockIdx.y * 16, n0 = blockIdx.x * 64;
  int lane = threadIdx.x, half = lane >> 4, li = lane & 15;
  int mr  = m0 + li;
  int mrc = (mr < M) ? mr : (M - 1);           // rows >= M never stored
  const float* wa = W + (size_t)mrc * K + 2 * half;

  const float* xq0 = Xb + (size_t)(2 * half) * N + (n0 + li);
  const float* xq1 = xq0 + 16;
  const float* xq2 = xq0 + 32;
  const float* xq3 = xq0 + 48;

  v8f acc[4] = {};
  const int K4 = K & ~3;
  for (int k0 = 0; k0 < K4; k0 += 4) {
    v2f a; a[0] = wa[0]; a[1] = wa[1];
    v2f q0; q0[0] = xq0[0]; q0[1] = xq0[N];
    v2f q1; q1[0] = xq1[0]; q1[1] = xq1[N];
    v2f q2; q2[0] = xq2[0]; q2[1] = xq2[N];
    v2f q3; q3[0] = xq3[0]; q3[1] = xq3[N];
    acc[0] = wmma_f32_k4(a, q0, acc[0]);
    acc[1] = wmma_f32_k4(a, q1, acc[1]);
    acc[2] = wmma_f32_k4(a, q2, acc[2]);
    acc[3] = wmma_f32_k4(a, q3, acc[3]);
    wa += 4;
    xq0 += (size_t)4 * N; xq1 += (size_t)4 * N;
    xq2 += (size_t)4 * N; xq3 += (size_t)4 * N;
  }
  if (K4 < K) {
    int ka = K4 + 2 * half;
    int c0 = (ka     < K) ? ka     : K - 1;
    int c1 = (ka + 1 < K) ? ka + 1 : K - 1;
    bool ok0 = ka < K, ok1 = (ka + 1) < K;
    v2f a;
    a[0] = W[(size_t)mrc * K + c0];
    a[1] = W[(size_t)mrc * K + c1];
    const float* r0 = Xb + (size_t)c0 * N + (n0 + li);
    const float* r1 = Xb + (size_t)c1 * N + (n0 + li);
#pragma unroll
    for (int t = 0; t < 4; ++t) {
      float v0 = r0[t * 16], v1 = r1[t * 16];
      v2f bf;
      bf[0] = ok0 ? v0 : 0.0f;
      bf[1] = ok1 ? v1 : 0.0f;
      acc[t] = wmma_f32_k4(a, bf, acc[t]);
    }
  }

#pragma unroll
  for (int t = 0; t < 4; ++t) {
    int col = n0 + t * 16 + li;
#pragma unroll
    for (int j = 0; j < 8; ++j) {
      int m = m0 + j + 8 * half;
      if (m < M) {
        float v = acc[t][j];
        if (gam) v *= RSQ * gam[m];
        v += bet[m];
        if (relu) v = fmaxf(v, 0.0f);
        Yb[(size_t)m * N + col] = v;
      }
    }
  }
}

// max over the 32 samples of each group: buf[b][c][n][32] -> feat[b][c][n]
__global__ void k_maxpool(const float* __restrict__ buf, float* __restrict__ feat,
                          int C, int np) {
  int t = blockIdx.x * blockDim.x + threadIdx.x;
  if (t >= B_ * C * np) return;
  const float* p = buf + (size_t)t * 32;
  float m = p[0];
  for (int s = 1; s < 32; ++s) m = fmaxf(m, p[s]);
  feat[t] = m;
}

// ---------------------------------------------------------------------------
// 3-NN (lax.top_k tie-break: earliest index) + inverse-distance weights
// ---------------------------------------------------------------------------
__global__ void k_threenn(const float* __restrict__ ux, const float* __restrict__ kx,
                          int Nu, int Nk, int* __restrict__ idx3, float* __restrict__ w3) {
  int t = blockIdx.x * blockDim.x + threadIdx.x;
  if (t >= B_ * Nu) return;
  int b = t / Nu;
  const float* q = ux + (size_t)t * 3;
  float qx = q[0], qy = q[1], qz = q[2];
  const float* x = kx + (size_t)b * Nk * 3;
  float d0 = 3.4e38f, d1 = 3.4e38f, d2 = 3.4e38f;
  int   i0 = 0, i1 = 0, i2 = 0;
  for (int i = 0; i < Nk; ++i) {
    float dx = x[i * 3 + 0] - qx, dy = x[i * 3 + 1] - qy, dz = x[i * 3 + 2] - qz;
    float d = dx * dx + dy * dy + dz * dz;
    if (d < d0)      { d2 = d1; i2 = i1; d1 = d0; i1 = i0; d0 = d; i0 = i; }
    else if (d < d1) { d2 = d1; i2 = i1; d1 = d;  i1 = i; }
    else if (d < d2) { d2 = d;  i2 = i; }
  }
  float r0 = 1.0f / (sqrtf(fmaxf(d0, 0.0f)) + 1e-8f);
  float r1 = 1.0f / (sqrtf(fmaxf(d1, 0.0f)) + 1e-8f);
  float r2 = 1.0f / (sqrtf(fmaxf(d2, 0.0f)) + 1e-8f);
  float s = r0 + r1 + r2;
  idx3[t * 3 + 0] = i0; idx3[t * 3 + 1] = i1; idx3[t * 3 + 2] = i2;
  w3[t * 3 + 0] = r0 / s; w3[t * 3 + 1] = r1 / s; w3[t * 3 + 2] = r2 / s;
}

// concat([3-NN interpolated featk, featc]) along channels -> h[b][Ck+Cc][Nu]
__global__ void k_fp_concat(const float* __restrict__ featk, const float* __restrict__ featc,
                            const int* __restrict__ idx3, const float* __restrict__ w3,
                            int Ck, int Nk, int Cc, int Nu, float* __restrict__ h) {
  int Ct = Ck + Cc;
  int t = blockIdx.x * blockDim.x + threadIdx.x;
  if (t >= B_ * Ct * Nu) return;
  int n = t % Nu;
  int c = (t / Nu) % Ct;
  int b = t / (Nu * Ct);
  float v;
  if (c < Ck) {
    size_t base = ((size_t)b * Nu + n) * 3;
    const float* f = featk + ((size_t)b * Ck + c) * Nk;
    v = f[idx3[base + 0]] * w3[base + 0] +
        f[idx3[base + 1]] * w3[base + 1] +
        f[idx3[base + 2]] * w3[base + 2];
  } else {
    v = featc[((size_t)b * Cc + (c - Ck)) * Nu + n];
  }
  h[t] = v;
}

// (B,21,P) -> (B,P,21)
__global__ void k_out_transpose(const float* __restrict__ y, float* __restrict__ out) {
  int t = blockIdx.x * blockDim.x + threadIdx.x;
  if (t >= B_ * P_ * NCLS) return;
  int o = t % NCLS;
  int p = (t / NCLS) % P_;
  int b = t / (NCLS * P_);
  out[t] = y[((size_t)b * NCLS + o) * P_ + p];
}

// ---------------------------------------------------------------------------
// Host orchestration
// ---------------------------------------------------------------------------
extern "C" void kernel_launch(void* const* d_in, const int* in_sizes, int n_in,
                              void* d_out, int out_size, void* d_ws, size_t ws_size,
                              hipStream_t stream) {
  (void)n_in; (void)out_size; (void)ws_size;
  const float* pc  = (const float*)d_in[0];   // (B,P,3)
  const float* img = (const float*)d_in[1];   // (B*V,128,32,41)
  const int*   i3  = (const int*)d_in[2];     // (B*V,P+1)
  const int*   i2  = (const int*)d_in[3];     // (B*V,P+1)

  struct Layer { const float* W; const float* g; const float* b; int M; int K; };
  Layer sa[4][3], fp[4][3], fc1;
  const float *fc2W, *fc2b;
  const int fpn[4]      = {3, 2, 2, 2};
  const int saM[4][3]   = {{32,32,64},{64,64,128},{128,128,256},{256,256,512}};
  const int saK[4][3]   = {{131,32,32},{67,64,64},{131,128,128},{259,256,256}};
  const int fpM[4][3]   = {{128,128,128},{256,128,0},{256,256,0},{256,256,0}};
  const int fpK[4][3]   = {{256,128,128},{320,256,0},{384,256,0},{768,256,0}};

  int cur = 4;
  auto take3 = [&](Layer& L, int M, int K) {
    L.W = (const float*)d_in[cur]; L.g = (const float*)d_in[cur+1];
    L.b = (const float*)d_in[cur+2]; L.M = M; L.K = K; cur += 3;
  };
  // params flattening: insertion ('sa','fp','fc1','fc2') vs jax-sorted
  // ('fc1','fc2','fp','sa'). sa[0][0].W has 32*131=4192 elems, fc1.W 16384.
  bool sa_first = (in_sizes[4] == 32 * 131);
  if (sa_first) {
    for (int l = 0; l < 4; ++l) for (int j = 0; j < 3; ++j) take3(sa[l][j], saM[l][j], saK[l][j]);
    for (int l = 0; l < 4; ++l) for (int j = 0; j < fpn[l]; ++j) take3(fp[l][j], fpM[l][j], fpK[l][j]);
    take3(fc1, 128, 128);
    fc2W = (const float*)d_in[cur]; fc2b = (const float*)d_in[cur + 1]; cur += 2;
  } else {
    take3(fc1, 128, 128);
    fc2W = (const float*)d_in[cur]; fc2b = (const float*)d_in[cur + 1]; cur += 2;
    for (int l = 0; l < 4; ++l) for (int j = 0; j < fpn[l]; ++j) take3(fp[l][j], fpM[l][j], fpK[l][j]);
    for (int l = 0; l < 4; ++l) for (int j = 0; j < 3; ++j) take3(sa[l][j], saM[l][j], saK[l][j]);
  }

  // workspace carve
  char* ws = (char*)d_ws;
  size_t off = 0;
  auto carve = [&](size_t bytes) -> void* {
    void* p = ws + off; off += (bytes + 255) & ~(size_t)255; return p;
  };
  int*   src_of  = (int*)  carve((size_t)B_ * V_ * P_ * 4);
  float* lfeat0  = (float*)carve((size_t)B_ * 128 * P_ * 4);
  float* lxyz1   = (float*)carve((size_t)B_ * 1024 * 3 * 4);
  float* lxyz2   = (float*)carve((size_t)B_ * 256 * 3 * 4);
  float* lxyz3   = (float*)carve((size_t)B_ * 64 * 3 * 4);
  float* lxyz4   = (float*)carve((size_t)B_ * 16 * 3 * 4);
  float* lfeat1  = (float*)carve((size_t)B_ * 128 * 1024 * 4);
  float* lfeat2  = (float*)carve((size_t)B_ * 256 * 256 * 4);
  float* lfeat3  = (float*)carve((size_t)B_ * 256 * 64 * 4);
  float* lfeat4  = (float*)carve((size_t)B_ * 512 * 16 * 4);
  int*   fpsidx  = (int*)  carve((size_t)B_ * 1024 * 4);
  int*   ballidx = (int*)  carve((size_t)B_ * 1024 * 32 * 4);
  int*   idx3    = (int*)  carve((size_t)B_ * P_ * 3 * 4);
  float* w3      = (float*)carve((size_t)B_ * P_ * 3 * 4);
  float* bufA    = (float*)carve((size_t)B_ * 64 * 32768 * 4);   // 64 MB
  float* bufB    = (float*)carve((size_t)B_ * 32 * 32768 * 4);   // 32 MB

  // ---- projection + view max ----
  { int tot = B_ * V_ * P_;  k_scatter<<<(tot + 255) / 256, 256, 0, stream>>>(i3, i2, src_of); }
  { int tot = B_ * CIMG * P_; k_imf<<<(tot + 255) / 256, 256, 0, stream>>>(img, src_of, lfeat0); }

  const float* xyz_lv[5]      = { pc, lxyz1, lxyz2, lxyz3, lxyz4 };
  float*       nxyz_lv[4]     = { lxyz1, lxyz2, lxyz3, lxyz4 };
  const float* feat_in_lv[4]  = { lfeat0, lfeat1, lfeat2, lfeat3 };
  float*       feat_out_lv[4] = { lfeat1, lfeat2, lfeat3, lfeat4 };
  const int    Nlv[5]         = { P_, 1024, 256, 64, 16 };
  const float  rad[4]         = { 0.1f, 0.2f, 0.4f, 0.8f };
  const int    Cf_lv[4]       = { 128, 64, 128, 256 };

  // ---- Set-Abstraction levels ----
  for (int L = 0; L < 4; ++L) {
    int N = Nlv[L], np = Nlv[L + 1];
    k_fps<<<B_, 256, 0, stream>>>(xyz_lv[L], N, np, fpsidx);
    { int tot = B_ * np; k_gather_xyz<<<(tot + 255) / 256, 256, 0, stream>>>(xyz_lv[L], fpsidx, N, np, nxyz_lv[L]); }
    { int tot = B_ * np; k_ball<<<(tot + 255) / 256, 256, 0, stream>>>(xyz_lv[L], nxyz_lv[L], N, np, rad[L] * rad[L], ballidx); }
    int Ncol = np * 32;
    { dim3 g(Ncol / 64, sa[L][0].M / 16, B_);
      k_sa_gemm1<<<g, 32, 0, stream>>>(xyz_lv[L], nxyz_lv[L], feat_in_lv[L], ballidx,
                                       sa[L][0].W, sa[L][0].g, sa[L][0].b, bufA,
                                       N, np, Cf_lv[L], sa[L][0].M); }
    { dim3 g(Ncol / 64, sa[L][1].M / 16, B_);
      k_gemm<<<g, 32, 0, stream>>>(bufA, sa[L][1].W, sa[L][1].g, sa[L][1].b, bufB,
                                   sa[L][1].M, Ncol, sa[L][1].K, 1); }
    { dim3 g(Ncol / 64, sa[L][2].M / 16, B_);
      k_gemm<<<g, 32, 0, stream>>>(bufB, sa[L][2].W, sa[L][2].g, sa[L][2].b, bufA,
                                   sa[L][2].M, Ncol, sa[L][2].K, 1); }
    { int tot = B_ * sa[L][2].M * np;
      k_maxpool<<<(tot + 255) / 256, 256, 0, stream>>>(bufA, feat_out_lv[L], sa[L][2].M, np); }
  }

  // ---- Feature-Propagation levels (i = 3 .. 0) ----
  float* featbuf[5] = { lfeat0, lfeat1, lfeat2, lfeat3, lfeat4 };
  int    curC[5]    = { 128, 64, 128, 256, 512 };
  for (int i = 3; i >= 0; --i) {
    int Nu = Nlv[i], Nk = Nlv[i + 1];
    int Ck = curC[i + 1], Cc = curC[i];
    { int tot = B_ * Nu; k_threenn<<<(tot + 255) / 256, 256, 0, stream>>>(xyz_lv[i], xyz_lv[i + 1], Nu, Nk, idx3, w3); }
    { int tot = B_ * (Ck + Cc) * Nu;
      k_fp_concat<<<(tot + 255) / 256, 256, 0, stream>>>(featbuf[i + 1], featbuf[i], idx3, w3,
                                                         Ck, Nk, Cc, Nu, bufA); }
    const float* Xcur = bufA;
    for (int j = 0; j < fpn[i]; ++j) {
      float* Yd = (j == fpn[i] - 1) ? featbuf[i] : ((j & 1) ? bufA : bufB);
      dim3 g(Nu / 64, fp[i][j].M / 16, B_);
      k_gemm<<<g, 32, 0, stream>>>(Xcur, fp[i][j].W, fp[i][j].g, fp[i][j].b, Yd,
                                   fp[i][j].M, Nu, fp[i][j].K, 1);
      Xcur = Yd;
    }
    curC[i] = fp[i][fpn[i] - 1].M;
  }

  // ---- head: fc1 (bn+relu) -> fc2 (bias only) -> transpose ----
  { dim3 g(P_ / 64, fc1.M / 16, B_);
    k_gemm<<<g, 32, 0, stream>>>(lfeat0, fc1.W, fc1.g, fc1.b, bufA, 128, P_, 128, 1); }
  { dim3 g(P_ / 64, (NCLS + 15) / 16, B_);
    k_gemm<<<g, 32, 0, stream>>>(bufA, fc2W, nullptr, fc2b, bufB, NCLS, P_, 128, 0); }
  { int tot = B_ * P_ * NCLS;
    k_out_transpose<<<(tot + 255) / 256, 256, 0, stream>>>(bufB, (float*)d_out); }
}